// Attention_matrix_62225486184557
// MI455X (gfx1250) — compile-verified
//
#include <hip/hip_runtime.h>
#include <hip/hip_bf16.h>

typedef float v2f __attribute__((ext_vector_type(2)));
typedef float v8f __attribute__((ext_vector_type(8)));

// Problem dims (fixed by the reference)
#define BDIM 8
#define NDIM 2048
#define MDIM 2048
#define DDIM 128

// Each wave computes a 16(N) x 64(M) output strip using V_WMMA_F32_16X16X4_F32.
// 8 waves per block -> block covers 128(N) x 64(M).
// Row norms are accumulated from the same fragments fed to WMMA (no extra traffic).
__global__ __launch_bounds__(256)
void pairdist_wmma_kernel(const float* __restrict__ Lp,
                          const float* __restrict__ Rp,
                          float* __restrict__ Out)
{
    const int lane = threadIdx.x & 31;
    const int wave = threadIdx.x >> 5;
    const int half = lane >> 4;    // 0: K=0..1, 1: K=2..3 within a 16x4 fragment
    const int lrow = lane & 15;    // fragment row (A) / fragment column-row (B)

    const int b  = blockIdx.z;
    const int n0 = (blockIdx.y * 8 + wave) * 16;
    const int m0 = blockIdx.x * 64;

    // A fragment base: left row (n0+lrow), starting column 2*half
    const float* __restrict__ Abase =
        Lp + ((size_t)b * NDIM + (size_t)(n0 + lrow)) * DDIM + 2 * half;
    // B fragment base for tile t: right row (m0 + t*16 + lrow), same column pattern
    const float* __restrict__ Bbase =
        Rp + ((size_t)b * MDIM + (size_t)(m0 + lrow)) * DDIM + 2 * half;

    v8f acc0 = {}, acc1 = {}, acc2 = {}, acc3 = {};
    float sa = 0.0f;
    float sb0 = 0.0f, sb1 = 0.0f, sb2 = 0.0f, sb3 = 0.0f;

#pragma unroll
    for (int k = 0; k < DDIM; k += 4) {
        v2f a = *(const v2f*)(Abase + k);
        sa = fmaf(a.x, a.x, sa);
        sa = fmaf(a.y, a.y, sa);

        v2f b0 = *(const v2f*)(Bbase + 0 * 16 * DDIM + k);
        v2f b1 = *(const v2f*)(Bbase + 1 * 16 * DDIM + k);
        v2f b2 = *(const v2f*)(Bbase + 2 * 16 * DDIM + k);
        v2f b3 = *(const v2f*)(Bbase + 3 * 16 * DDIM + k);

        sb0 = fmaf(b0.x, b0.x, sb0); sb0 = fmaf(b0.y, b0.y, sb0);
        sb1 = fmaf(b1.x, b1.x, sb1); sb1 = fmaf(b1.y, b1.y, sb1);
        sb2 = fmaf(b2.x, b2.x, sb2); sb2 = fmaf(b2.y, b2.y, sb2);
        sb3 = fmaf(b3.x, b3.x, sb3); sb3 = fmaf(b3.y, b3.y, sb3);

        // 8-arg form: (neg_a, A, neg_b, B, c_mod, C, reuse_a, reuse_b)
        acc0 = __builtin_amdgcn_wmma_f32_16x16x4_f32(false, a, false, b0, (short)0, acc0, false, false);
        acc1 = __builtin_amdgcn_wmma_f32_16x16x4_f32(false, a, false, b1, (short)0, acc1, false, false);
        acc2 = __builtin_amdgcn_wmma_f32_16x16x4_f32(false, a, false, b2, (short)0, acc2, false, false);
        acc3 = __builtin_amdgcn_wmma_f32_16x16x4_f32(false, a, false, b3, (short)0, acc3, false, false);
    }

    // Full row norms: lane L and L^16 each hold half of row (lrow)'s sum of squares.
    float l2 = sa + __shfl_xor(sa, 16, 32);         // lane -> ||left row (n0+lrow)||^2
    float r2t[4];
    r2t[0] = sb0 + __shfl_xor(sb0, 16, 32);         // lane -> ||right row (m0+t*16+lrow)||^2
    r2t[1] = sb1 + __shfl_xor(sb1, 16, 32);
    r2t[2] = sb2 + __shfl_xor(sb2, 16, 32);
    r2t[3] = sb3 + __shfl_xor(sb3, 16, 32);

    // Pre-gather l2 for each output row v + 8*half (one ds_bpermute per v).
    float l2row[8];
#pragma unroll
    for (int v = 0; v < 8; ++v) {
        l2row[v] = __shfl(l2, v + 8 * half, 32);
    }

    v8f accs[4] = {acc0, acc1, acc2, acc3};
    const size_t outRowBase = ((size_t)b * NDIM + (size_t)n0) * MDIM + (size_t)m0;

#pragma unroll
    for (int t = 0; t < 4; ++t) {
        // C/D layout: element (row = v + 8*half, col = lrow). The r2 term for
        // col == lrow is already in this lane's r2t[t].
        const size_t colBase = outRowBase + (size_t)(t * 16 + lrow);
#pragma unroll
        for (int v = 0; v < 8; ++v) {
            const int row = v + 8 * half;
            float d2 = fmaf(-2.0f, accs[t][v], l2row[v] + r2t[t]);
            d2 = fmaxf(d2, 0.0f);
            float den = 1.0f + __builtin_sqrtf(d2);
            den = fmaxf(den, 1e-7f);
            Out[colBase + (size_t)row * MDIM] = 1.0f / den;
        }
    }
}

extern "C" void kernel_launch(void* const* d_in, const int* in_sizes, int n_in,
                              void* d_out, int out_size, void* d_ws, size_t ws_size,
                              hipStream_t stream) {
    (void)in_sizes; (void)n_in; (void)out_size; (void)d_ws; (void)ws_size;
    const float* left  = (const float*)d_in[0];  // [B, N, D]
    const float* right = (const float*)d_in[1];  // [B, M, D]
    float* out = (float*)d_out;                  // [B, N, M]

    dim3 block(256, 1, 1);
    dim3 grid(MDIM / 64, NDIM / 128, BDIM);
    pairdist_wmma_kernel<<<grid, block, 0, stream>>>(left, right, out);
}